// DPLookupRenderer_11192684773862
// MI455X (gfx1250) — compile-verified
//
#include <hip/hip_runtime.h>

// DPLookupRenderer: part-indexed bilinear texture gather.
// B=4, C=16, P=128 (atlas 4x6 patches -> 512x768), H=W=1024.
//
// Two-phase plan (workspace permitting):
//  1) dp_transpose_kernel: re-layout dp_comb [B,C,512,768] -> tex[b][k][x][y][c]
//     (channels innermost, 64B per tap). LDS-tiled so loads AND stores are fully
//     coalesced. dp is NT-loaded (never needed again); tex stores are RT so the
//     100.7MB tex array stays resident in the 192MB L2.
//  2) dp_gather_kernel: 1 thread/pixel. 3 NT scalar loads (iuv), 4 taps x 4
//     b128 loads (each tap = contiguous 64B -> ~3-4 cache lines per pixel
//     instead of ~32 with the original layout), 16 NT b32 stores.
// Steady-state HBM ~500MB (~21us @ 23.3TB/s); L2 gather fill ~1.5-2GB vs ~16GB
// for the naive layout.

typedef float v4f __attribute__((ext_vector_type(4)));

#define B_    4
#define C_    16
#define P_    128
#define H_    1024
#define W_    1024
#define AW_   (6 * P_)           // 768  atlas width
#define AH_   (4 * P_)           // 512  atlas height
#define CHS_  (AH_ * AW_)        // 393216 channel stride in dp_comb
#define HW_   (H_ * W_)          // 2^20
#define KPC_  (P_ * P_ * C_)     // 262144 floats per (b,k) patch in tex

// ---------------------------------------------------------------------------
// Phase 1: [B,C,512,768] -> tex[(((b*24+k)*128+x)*128+y)*16+c]
// One block per (b,k,x): reads a 16(c) x 128(y) tile, writes it transposed.
// ---------------------------------------------------------------------------
__global__ __launch_bounds__(256)
void dp_transpose_kernel(const float* __restrict__ dp, float* __restrict__ tex)
{
    __shared__ float lds[C_ * 132];          // pad 132: conflict-free transpose

    const int bid = blockIdx.x;              // b*24*128 + k*128 + x
    const int X   = bid & (P_ - 1);
    const int bk  = bid >> 7;                // b*24 + k
    const int k   = bk % 24;
    const int b   = bk / 24;
    const int row = k / 6;
    const int col = k - row * 6;
    const int t   = threadIdx.x;

    // src row for channel c: dp[b][c][row*128 + X][col*128 + y], y contiguous
    const float* src = dp + (size_t)b * C_ * CHS_
                          + (size_t)(row * P_ + X) * AW_ + col * P_;
#pragma unroll
    for (int it = 0; it < 8; ++it) {         // 2048 floats / 256 threads
        const int idx = t + it * 256;
        const int c = idx >> 7, y = idx & (P_ - 1);
        lds[c * 132 + y] = __builtin_nontemporal_load(src + (size_t)c * CHS_ + y);
    }
    __syncthreads();

    float* dst = tex + (size_t)bk * KPC_ + (size_t)X * (P_ * C_);
#pragma unroll
    for (int it = 0; it < 8; ++it) {
        const int idx = t + it * 256;        // idx = y*16 + c, contiguous writes
        const int y = idx >> 4, c = idx & (C_ - 1);
        dst[idx] = lds[c * 132 + y];         // RT store: keep tex hot in L2
    }
}

// ---------------------------------------------------------------------------
// Phase 2: bilinear gather from channel-last tex. 1 thread = 1 pixel.
// ---------------------------------------------------------------------------
__global__ __launch_bounds__(256)
void dp_gather_kernel(const float* __restrict__ tex,
                      const float* __restrict__ iuv,
                      float* __restrict__ out)
{
    const int i   = blockIdx.x * 256 + threadIdx.x;  // pixel id < 4M
    const int b   = i >> 20;                         // HW_ == 2^20
    const int pix = i & (HW_ - 1);

    const float* iuvb = iuv + (size_t)b * 3 * HW_;
    const float pidf = __builtin_nontemporal_load(iuvb + pix);
    const float u    = __builtin_nontemporal_load(iuvb + HW_ + pix);
    const float v    = __builtin_nontemporal_load(iuvb + 2 * HW_ + pix);

    const int  pi    = __float2int_rn(pidf);         // exact-integer part ids
    const bool valid = (pi >= 1) && (pi <= 24);
    const int  k     = min(max(pi - 1, 0), 23);

    // align_corners=True mapping into the 128x128 patch
    const float fx  = (u + 1.0f) * 0.5f * (float)(P_ - 1);  // x axis
    const float fy  = (v + 1.0f) * 0.5f * (float)(P_ - 1);
    const float x0f = floorf(fx), y0f = floorf(fy);
    const float wx  = fx - x0f,   wy  = fy - y0f;
    const int   x0  = (int)x0f,   y0  = (int)y0f;

    const float* texb = tex + ((size_t)b * 24 + k) * KPC_;

    v4f acc[4];
#pragma unroll
    for (int q = 0; q < 4; ++q) acc[q] = (v4f){0.f, 0.f, 0.f, 0.f};

#pragma unroll
    for (int s = 0; s < 4; ++s) {            // (0,0)(1,0)(0,1)(1,1) = ref order
        const int  dx  = s & 1, dy = s >> 1;
        const int  xi  = x0 + dx, yi = y0 + dy;
        const bool inb = valid && (xi >= 0) && (xi < P_) && (yi >= 0) && (yi < P_);
        const int  xc  = min(max(xi, 0), P_ - 1);
        const int  yc  = min(max(yi, 0), P_ - 1);
        const float w  = inb ? ((dx ? wx : 1.0f - wx) * (dy ? wy : 1.0f - wy))
                             : 0.0f;         // zero-padding + validity folded in
        const float* tp = texb + ((size_t)(xc * P_ + yc)) * C_;  // 64B aligned
#pragma unroll
        for (int q = 0; q < 4; ++q) {
            const v4f tv = *(const v4f*)(tp + 4 * q);            // b128 load
            acc[q] += tv * w;
        }
    }

    float* ob = out + ((size_t)b * C_) * HW_ + pix;
#pragma unroll
    for (int q = 0; q < 4; ++q)
#pragma unroll
        for (int r = 0; r < 4; ++r)
            __builtin_nontemporal_store(acc[q][r],
                                        ob + (size_t)(q * 4 + r) * HW_);
}

// ---------------------------------------------------------------------------
// Fallback: direct gather from original layout (if workspace is too small).
// ---------------------------------------------------------------------------
__global__ __launch_bounds__(256)
void dp_lookup_direct(const float* __restrict__ dp,
                      const float* __restrict__ iuv,
                      float* __restrict__ out)
{
    const int t    = blockIdx.x * 256 + threadIdx.x;
    const int quad = t << 2;
    const int b    = quad >> 20;
    const int pix  = quad & (HW_ - 1);

    const float* iuvb = iuv + (size_t)b * 3 * HW_;
    const v4f pidv = __builtin_nontemporal_load((const v4f*)(iuvb + pix));
    const v4f uv   = __builtin_nontemporal_load((const v4f*)(iuvb + HW_ + pix));
    const v4f vv   = __builtin_nontemporal_load((const v4f*)(iuvb + 2 * HW_ + pix));

    int   off[4][4];
    float wt[4][4];
#pragma unroll
    for (int j = 0; j < 4; ++j) {
        const int   pi    = __float2int_rn(pidv[j]);
        const bool  valid = (pi >= 1) && (pi <= 24);
        const int   k     = min(max(pi - 1, 0), 23);
        const int   row   = k / 6;
        const int   col   = k - row * 6;
        const float fx  = (uv[j] + 1.0f) * 0.5f * (float)(P_ - 1);
        const float fy  = (vv[j] + 1.0f) * 0.5f * (float)(P_ - 1);
        const float x0f = floorf(fx), y0f = floorf(fy);
        const float wx  = fx - x0f,   wy  = fy - y0f;
        const int   x0  = (int)x0f,   y0  = (int)y0f;
#pragma unroll
        for (int s = 0; s < 4; ++s) {
            const int  dx  = s & 1, dy = s >> 1;
            const int  xi  = x0 + dx, yi = y0 + dy;
            const bool inb = valid && (xi >= 0) && (xi < P_) && (yi >= 0) && (yi < P_);
            const int  xc  = min(max(xi, 0), P_ - 1);
            const int  yc  = min(max(yi, 0), P_ - 1);
            off[j][s] = (row * P_ + xc) * AW_ + (col * P_ + yc);
            wt[j][s]  = inb ? ((dx ? wx : 1.0f - wx) * (dy ? wy : 1.0f - wy)) : 0.0f;
        }
    }

    const float* dpb  = dp  + (size_t)b * C_ * CHS_;
    float*       outb = out + (size_t)b * C_ * HW_ + pix;
#pragma unroll
    for (int c = 0; c < C_; ++c) {
        const float* dpc = dpb + (size_t)c * CHS_;
        v4f r;
#pragma unroll
        for (int j = 0; j < 4; ++j) {
            float acc =      dpc[off[j][0]] * wt[j][0];
            acc = fmaf(dpc[off[j][1]], wt[j][1], acc);
            acc = fmaf(dpc[off[j][2]], wt[j][2], acc);
            acc = fmaf(dpc[off[j][3]], wt[j][3], acc);
            r[j] = acc;
        }
        __builtin_nontemporal_store(r, (v4f*)(outb + (size_t)c * HW_));
    }
}

extern "C" void kernel_launch(void* const* d_in, const int* in_sizes, int n_in,
                              void* d_out, int out_size, void* d_ws, size_t ws_size,
                              hipStream_t stream) {
    const float* dp  = (const float*)d_in[0];   // [4,16,512,768] fp32
    const float* iuv = (const float*)d_in[1];   // [4,3,1024,1024] fp32
    float*       out = (float*)d_out;           // [4,16,1024,1024] fp32

    const size_t tex_bytes = (size_t)B_ * 24 * KPC_ * sizeof(float); // 100.7MB

    if (ws_size >= tex_bytes) {
        float* tex = (float*)d_ws;
        // Phase 1: re-layout (one block per (b,k,x) row) -- stream-ordered.
        dp_transpose_kernel<<<B_ * 24 * P_, 256, 0, stream>>>(dp, tex);
        // Phase 2: gather (one thread per pixel).
        dp_gather_kernel<<<(B_ * HW_) / 256, 256, 0, stream>>>(tex, iuv, out);
    } else {
        dp_lookup_direct<<<(B_ * HW_ / 4) / 256, 256, 0, stream>>>(dp, iuv, out);
    }
}